// rnn_test_71382356459529
// MI455X (gfx1250) — compile-verified
//
#include <hip/hip_runtime.h>

#define Bn   128
#define Tn   500
#define INn  700
#define Hn   512
#define OUTn 20
#define BRn  4
#define Kn   (INn + Hn)   // 1212
#define Jn   (Hn * BRn)   // 2048
#define VTH  0.5f
#define KC   32           // K chunk

typedef __attribute__((ext_vector_type(2))) float v2f;
typedef __attribute__((ext_vector_type(8))) float v8f;
typedef __attribute__((ext_vector_type(4))) int   v4i;
typedef __attribute__((address_space(1))) v4i gv4i;   // global int4
typedef __attribute__((address_space(3))) v4i lv4i;   // LDS int4

// ---- async load-to-LDS availability (compile-safe fallback) ----
#if defined(__has_builtin)
#  if __has_builtin(__builtin_amdgcn_global_load_async_to_lds_b128)
#    define USE_ASYNC 1
#  endif
#endif
#ifndef USE_ASYNC
#  define USE_ASYNC 0
#endif

__device__ __forceinline__ void async_wait0() {
#if USE_ASYNC
#  if __has_builtin(__builtin_amdgcn_s_wait_asynccnt)
  __builtin_amdgcn_s_wait_asynccnt(0);
#  else
  asm volatile("s_wait_asynccnt 0x0" ::: "memory");
#  endif
#endif
}

__device__ __forceinline__ float sigm(float v) { return 1.0f / (1.0f + __expf(-v)); }

// ---------------- prologue kernels ----------------

__global__ void k_zero(float* __restrict__ p, int n) {
  int i = blockIdx.x * blockDim.x + threadIdx.x;
  if (i < n) p[i] = 0.0f;
}

// WmT[k][j] = branch_mask[j][k] ? W1[j][k] : 0   (transpose + mask)
__global__ void k_build_wmt(const float* __restrict__ W1,
                            const unsigned char* __restrict__ bmask,
                            float* __restrict__ WmT) {
  int idx = blockIdx.x * blockDim.x + threadIdx.x;
  if (idx >= Kn * Jn) return;
  int i = idx / Jn;          // k  (0..1211)
  int j = idx - i * Jn;      // j  (0..2047)
  float w = W1[(size_t)j * Kn + i];
  WmT[idx] = bmask[(size_t)j * Kn + i] ? w : 0.0f;
}

// ---------------- per-timestep hidden layer (WMMA f32, 2x2 register tiling) --
// grid (B/32, Jn/128), block 128 (4 waves). Each wave: 32(b) x 32(j) = 4 tiles.
__global__ __launch_bounds__(128)
void k_hidden(const float* __restrict__ x, const float* __restrict__ WmT,
              const float* __restrict__ b1, const float* __restrict__ tau_m1,
              const float* __restrict__ tau_n1, const unsigned char* __restrict__ rhy,
              const float* __restrict__ spk_prev, float* __restrict__ spk_next,
              float* __restrict__ mem1, float* __restrict__ dcur, int t)
{
  __shared__ float As[32][KC + 1];            // +1 pad: column reads conflict-free
  __shared__ __align__(16) float Bs[KC][128];

  const int tid  = threadIdx.x;
  const int lane = tid & 31;
  const int wave = tid >> 5;
  const int b0   = blockIdx.x * 32;
  const int j0   = blockIdx.y * 128;
  const int jw   = wave * 32;                 // wave's 32-wide j slice within block

  v8f acc00 = {}, acc01 = {}, acc10 = {}, acc11 = {};

  for (int kk = 0; kk < Kn; kk += KC) {
    // A tile: 32 rows (batch) x 32 cols (k) = concat(x_t, spk_prev), zero-padded
#pragma unroll
    for (int i = 0; i < 8; ++i) {
      int idx = tid + i * 128;     // 0..1023
      int r   = idx >> 5;          // 0..31
      int c   = idx & 31;
      int kg  = kk + c;
      float v = 0.0f;
      if (kg < INn)     v = x[(size_t)(b0 + r) * (Tn * INn) + (size_t)t * INn + kg];
      else if (kg < Kn) v = spk_prev[(b0 + r) * Hn + (kg - INn)];
      As[r][c] = v;
    }
    // B tile: 32 rows (k) x 128 cols (j) from L2-resident WmT
#if USE_ASYNC
#pragma unroll
    for (int i = 0; i < 8; ++i) {
      int f16 = tid + i * 128;     // 0..1023 sixteen-byte units
      int c   = f16 >> 5;          // k row (32 x 16B per 512B row)
      int jq  = (f16 & 31) << 2;   // j = 0,4,...,124
      int kg  = kk + c;
      if (kg < Kn) {
        __builtin_amdgcn_global_load_async_to_lds_b128(
            (gv4i*)(float*)(WmT + (size_t)kg * Jn + (j0 + jq)),
            (lv4i*)&Bs[c][jq],
            0, 0);
      } else {
        Bs[c][jq] = 0.0f; Bs[c][jq + 1] = 0.0f;
        Bs[c][jq + 2] = 0.0f; Bs[c][jq + 3] = 0.0f;
      }
    }
    async_wait0();
#else
#pragma unroll
    for (int i = 0; i < 32; ++i) {
      int idx = tid + i * 128;     // 0..4095
      int c   = idx >> 7;
      int j   = idx & 127;
      int kg  = kk + c;
      Bs[c][j] = (kg < Kn) ? WmT[(size_t)kg * Jn + (j0 + j)] : 0.0f;
    }
#endif
    __syncthreads();

    const int n16   = lane & 15;
    const int khalf = (lane < 16) ? 0 : 2;   // f32 frag: lanes 16-31 hold K+2,K+3
#pragma unroll
    for (int kb = 0; kb < KC / 4; ++kb) {
      int kc = kb * 4 + khalf;
      v2f a0, a1, f0, f1;
      a0.x = As[n16][kc];          a0.y = As[n16][kc + 1];
      a1.x = As[16 + n16][kc];     a1.y = As[16 + n16][kc + 1];
      f0.x = Bs[kc][jw + n16];     f0.y = Bs[kc + 1][jw + n16];
      f1.x = Bs[kc][jw + 16 + n16];f1.y = Bs[kc + 1][jw + 16 + n16];
      acc00 = __builtin_amdgcn_wmma_f32_16x16x4_f32(false, a0, false, f0, (short)0, acc00, false, false);
      acc01 = __builtin_amdgcn_wmma_f32_16x16x4_f32(false, a0, false, f1, (short)0, acc01, false, false);
      acc10 = __builtin_amdgcn_wmma_f32_16x16x4_f32(false, a1, false, f0, (short)0, acc10, false, false);
      acc11 = __builtin_amdgcn_wmma_f32_16x16x4_f32(false, a1, false, f1, (short)0, acc11, false, false);
    }
    __syncthreads();
  }

  // epilogue: C layout -> element (m = v + (lane<16?0:8), n = lane&15)
  const int n16  = lane & 15;
  const int mofs = (lane < 16) ? 0 : 8;
  v8f accs[2][2];
  accs[0][0] = acc00; accs[0][1] = acc01;
  accs[1][0] = acc10; accs[1][1] = acc11;

#pragma unroll
  for (int mt = 0; mt < 2; ++mt) {
#pragma unroll
    for (int nt = 0; nt < 2; ++nt) {
      const int j = j0 + jw + nt * 16 + n16;
      const int h = j >> 2;
      const float beta  = sigm(tau_n1[j]);
      const float alpha = sigm(tau_m1[h]);
      const float bias  = b1[j];
      const int   rm    = rhy[t * Hn + h];
      v8f acc = accs[mt][nt];
#pragma unroll
      for (int v = 0; v < 8; ++v) {
        int b = b0 + mt * 16 + v + mofs;
        float din = acc[v] + bias;
        float dc  = beta * dcur[b * Jn + j] + (1.0f - beta) * din;
        dcur[b * Jn + j] = dc;
        // l_in: sum over 4 branches = 4 adjacent lanes (same half, same row)
        float l = dc + __shfl_xor(dc, 1, 32);
        l += __shfl_xor(l, 2, 32);
        if ((lane & 3) == 0) {
          float mo = mem1[b * Hn + h];
          float so = spk_prev[b * Hn + h];
          float mn = alpha * mo + (1.0f - alpha) * l - VTH * so;
          float m  = rm ? mn : mo;
          mem1[b * Hn + h]     = m;
          spk_next[b * Hn + h] = (rm && (m - VTH > 0.0f)) ? 1.0f : 0.0f;
        }
      }
    }
  }
}

// ---------------- per-timestep output layer + log_softmax ----------------
// one wave32 per batch row; lanes 0..19 own one class each
__global__ __launch_bounds__(32)
void k_out(const float* __restrict__ spk1, const float* __restrict__ W2,
           const float* __restrict__ b2, const float* __restrict__ tau_m2,
           float* __restrict__ mem2, float* __restrict__ spk2,
           float* __restrict__ out, int t)
{
  const int b    = blockIdx.x;
  const int lane = threadIdx.x;
  float mn = -__builtin_inff();
  if (lane < OUTn) {
    float acc = b2[lane];
    const float* w = W2 + lane * Hn;
    const float* s = spk1 + b * Hn;
    for (int h = 0; h < Hn; ++h) acc += s[h] * w[h];
    float a2 = sigm(tau_m2[lane]);
    float mo = mem2[b * OUTn + lane];
    float so = spk2[b * OUTn + lane];
    mn = a2 * mo + (1.0f - a2) * acc - VTH * so;
    mem2[b * OUTn + lane] = mn;
    spk2[b * OUTn + lane] = (mn - VTH > 0.0f) ? 1.0f : 0.0f;
  }
  float mx = mn;
#pragma unroll
  for (int o = 16; o > 0; o >>= 1) mx = fmaxf(mx, __shfl_xor(mx, o, 32));
  float e = (lane < OUTn) ? __expf(mn - mx) : 0.0f;
  float sum = e;
#pragma unroll
  for (int o = 16; o > 0; o >>= 1) sum += __shfl_xor(sum, o, 32);
  if (lane < OUTn)
    out[(size_t)b * OUTn * Tn + (size_t)lane * Tn + t] = mn - mx - __logf(sum);
}

// ---------------- launcher ----------------
extern "C" void kernel_launch(void* const* d_in, const int* in_sizes, int n_in,
                              void* d_out, int out_size, void* d_ws, size_t ws_size,
                              hipStream_t stream)
{
  (void)in_sizes; (void)n_in; (void)out_size; (void)ws_size;
  const float* x       = (const float*)d_in[0];
  const float* W1      = (const float*)d_in[1];
  const float* b1      = (const float*)d_in[2];
  const float* tau_m1  = (const float*)d_in[3];
  const float* tau_n1  = (const float*)d_in[4];
  const float* W2      = (const float*)d_in[5];
  const float* b2      = (const float*)d_in[6];
  const float* tau_m2  = (const float*)d_in[7];
  const unsigned char* rhy   = (const unsigned char*)d_in[8];
  const unsigned char* bmask = (const unsigned char*)d_in[9];
  float* out = (float*)d_out;

  // workspace layout (floats): WmT | dcur | mem1 | spkA | spkB | mem2 | spk2  (~11.8 MB)
  float* ws   = (float*)d_ws;
  float* WmT  = ws;
  float* dcur = WmT  + (size_t)Kn * Jn;
  float* mem1 = dcur + (size_t)Bn * Jn;
  float* spkA = mem1 + (size_t)Bn * Hn;
  float* spkB = spkA + (size_t)Bn * Hn;
  float* mem2 = spkB + (size_t)Bn * Hn;
  float* spk2 = mem2 + (size_t)Bn * OUTn;

  {
    int n = Bn * Jn + 3 * Bn * Hn + 2 * Bn * OUTn;   // dcur..spk2 contiguous
    hipLaunchKernelGGL(k_zero, dim3((n + 255) / 256), dim3(256), 0, stream, dcur, n);
  }
  {
    int n = Kn * Jn;
    hipLaunchKernelGGL(k_build_wmt, dim3((n + 255) / 256), dim3(256), 0, stream,
                       W1, bmask, WmT);
  }

  for (int t = 0; t < Tn; ++t) {
    const float* sp = (t & 1) ? spkB : spkA;
    float*       sn = (t & 1) ? spkA : spkB;
    hipLaunchKernelGGL(k_hidden, dim3(Bn / 32, Jn / 128), dim3(128), 0, stream,
                       x, WmT, b1, tau_m1, tau_n1, rhy, sp, sn, mem1, dcur, t);
    hipLaunchKernelGGL(k_out, dim3(Bn), dim3(32), 0, stream,
                       sn, W2, b2, tau_m2, mem2, spk2, out, t);
  }
}